// Exprnn_10746008175150
// MI455X (gfx1250) — compile-verified
//
#include <hip/hip_runtime.h>
#include <hip/hip_bf16.h>

typedef __attribute__((ext_vector_type(2))) float v2f;
typedef __attribute__((ext_vector_type(8))) float v8f;

#define RNN_B 8192
#define RNN_T 2048

// ---------------------------------------------------------------------------
// Kernel 1: Bmat = expm(triu(A,1) - triu(A,1)^T) for a 10x10 matrix.
// Scaling-and-squaring + Taylor series. One block, tiny work.
// ---------------------------------------------------------------------------
__global__ void expm_skew10_kernel(const float* __restrict__ A,
                                   float* __restrict__ Bmat) {
    __shared__ float M[100], P[100], Pn[100], E[100];
    __shared__ int ssq;
    const int tid = threadIdx.x;
    const int i = tid / 10, j = tid % 10;

    if (tid < 100) {
        float u  = (j > i) ? A[i * 10 + j] : 0.f;   // triu(A,1)
        float ut = (i > j) ? A[j * 10 + i] : 0.f;   // triu(A,1)^T
        M[tid] = u - ut;                            // skew-symmetric
    }
    __syncthreads();

    if (tid == 0) {
        float nrm = 0.f;
        for (int r = 0; r < 10; ++r) {
            float rs = 0.f;
            for (int c = 0; c < 10; ++c) rs += fabsf(M[r * 10 + c]);
            nrm = fmaxf(nrm, rs);
        }
        int sq = 0;
        while (nrm > 0.25f && sq < 30) { nrm *= 0.5f; ++sq; }
        ssq = sq;
    }
    __syncthreads();

    const int sq = ssq;
    const float scale = exp2f((float)(-sq));
    if (tid < 100) {
        float m = M[tid] * scale;
        M[tid] = m;
        P[tid] = m;
        E[tid] = m + ((i == j) ? 1.f : 0.f);        // I + M
    }
    __syncthreads();

    // Taylor: E = sum_{k} M^k / k!
    for (int k = 2; k <= 18; ++k) {
        float acc = 0.f;
        if (tid < 100) {
            for (int t = 0; t < 10; ++t) acc += P[i * 10 + t] * M[t * 10 + j];
            acc /= (float)k;
        }
        __syncthreads();
        if (tid < 100) { P[tid] = acc; Pn[tid] = acc; E[tid] += acc; }
        __syncthreads();
    }
    // Squaring: E = E^(2^sq)
    for (int q = 0; q < sq; ++q) {
        float acc = 0.f;
        if (tid < 100) {
            for (int t = 0; t < 10; ++t) acc += E[i * 10 + t] * E[t * 10 + j];
        }
        __syncthreads();
        if (tid < 100) Pn[tid] = acc;
        __syncthreads();
        if (tid < 100) E[tid] = Pn[tid];
        __syncthreads();
    }
    if (tid < 100) Bmat[tid] = E[tid];
}

// ---------------------------------------------------------------------------
// Kernel 2: time-scan. One wave32 handles 16 batch rows.
//
// Computes D(16x16: j x b) = sum_k A(16x4) x B(4x16) via 3 chained
// v_wmma_f32_16x16x4_f32, where A = Bmat^T padded to K=12 with W_in in
// columns 10,11, and B rows 0..9 = h^T (recycled from previous D fragment
// via half-wave shuffles), rows 10,11 = raw inputs x0,x1.
// modrelu applied in the D fragment layout. Final linear head = one more
// 3-WMMA chain with lin_W tiles (bias folded via constant-1 B row).
// ---------------------------------------------------------------------------
__global__ __launch_bounds__(128) void exprnn_scan_kernel(
    const float* __restrict__ x,       // [B, T, 2]
    const float* __restrict__ Bmat,    // [10,10] (row-major, from d_ws)
    const float* __restrict__ W_in,    // [10,2]
    const float* __restrict__ b_mod,   // [10]
    const float* __restrict__ lin_W,   // [10,10]
    const float* __restrict__ lin_b,   // [10]
    float* __restrict__ out)           // [B,10]
{
    const int tid  = threadIdx.x;
    const int lane = tid & 31;
    const int half = lane >> 4;        // 0: lanes 0-15, 1: lanes 16-31
    const int lq   = lane & 15;
    const int wave = blockIdx.x * (blockDim.x >> 5) + (tid >> 5);
    const int m0   = wave * 16;        // first batch row of this wave

    // --- loop-invariant A fragments ---------------------------------------
    // A fragment layout (32-bit 16x4): vgpr v, lane -> (m = lq, k = v + 2*half)
    // WMMA jp covers global k = v + 2*half + 4*jp.
    v2f arec[3], aout[3];
#pragma unroll
    for (int jp = 0; jp < 3; ++jp) {
#pragma unroll
        for (int v = 0; v < 2; ++v) {
            const int k = v + 2 * half + 4 * jp;
            float ar = 0.f, ao = 0.f;
            if (lq < 10) {
                if (k < 10)       { ar = Bmat[k * 10 + lq];  ao = lin_W[lq * 10 + k]; }
                else if (k == 10) { ar = W_in[lq * 2 + 0];   ao = lin_b[lq]; }
                else              { ar = W_in[lq * 2 + 1];   ao = 0.f; }
            }
            arec[jp][v] = ar;
            aout[jp][v] = ao;
        }
    }

    // modrelu bias per D vgpr: D vgpr r, lane -> hidden j = r + 8*half
    float bm[8];
#pragma unroll
    for (int r = 0; r < 8; ++r) {
        const int jj = r + 8 * half;
        bm[r] = (jj < 10) ? b_mod[jj] : 0.f;
    }

    // Each lane streams inputs of batch row (m0 + lq); both halves read the
    // same address (coalesces/broadcasts in cache).
    const float2* __restrict__ xp =
        (const float2*)(x + (size_t)(m0 + lq) * RNN_T * 2);

    v8f h = {};  // h^T fragment, starts at zero

    for (int t = 0; t < RNN_T; ++t) {
        if ((t & 15) == 0)
            __builtin_prefetch((const void*)(xp + t + 64), 0, 0);
        const float2 xv = xp[t];

        // D fragment -> next B fragments (half-wave exchange).
        const float s2  = __shfl(h[2], lq, 32);
        const float s3  = __shfl(h[3], lq, 32);
        const float s6  = __shfl(h[6], lq, 32);
        const float s7  = __shfl(h[7], lq, 32);
        const float s0h = __shfl(h[0], lq + 16, 32);
        const float s1h = __shfl(h[1], lq + 16, 32);

        v2f b0, b1, b2;
        b0[0] = half ? s2 : h[0];     // k = 0 | 2
        b0[1] = half ? s3 : h[1];     // k = 1 | 3
        b1[0] = half ? s6 : h[4];     // k = 4 | 6
        b1[1] = half ? s7 : h[5];     // k = 5 | 7
        b2[0] = half ? xv.x : s0h;    // k = 8 | 10 (input x0)
        b2[1] = half ? xv.y : s1h;    // k = 9 | 11 (input x1)

        v8f c = {};
        c = __builtin_amdgcn_wmma_f32_16x16x4_f32(false, arec[0], false, b0,
                                                  (short)0, c, false, false);
        c = __builtin_amdgcn_wmma_f32_16x16x4_f32(false, arec[1], false, b1,
                                                  (short)0, c, false, false);
        c = __builtin_amdgcn_wmma_f32_16x16x4_f32(false, arec[2], false, b2,
                                                  (short)0, c, false, false);

        // modrelu: sign(z) * relu(|z| + b)
#pragma unroll
        for (int r = 0; r < 8; ++r) {
            const float z   = c[r];
            const float mag = fmaxf(fabsf(z) + bm[r], 0.f);
            h[r] = __builtin_copysignf(mag, z);
        }
    }

    // --- output head: out^T(o x b) = lin_W x h^T (+ bias via B row 10 = 1) --
    {
        const float s2  = __shfl(h[2], lq, 32);
        const float s3  = __shfl(h[3], lq, 32);
        const float s6  = __shfl(h[6], lq, 32);
        const float s7  = __shfl(h[7], lq, 32);
        const float s0h = __shfl(h[0], lq + 16, 32);
        const float s1h = __shfl(h[1], lq + 16, 32);

        v2f b0, b1, b2;
        b0[0] = half ? s2 : h[0];
        b0[1] = half ? s3 : h[1];
        b1[0] = half ? s6 : h[4];
        b1[1] = half ? s7 : h[5];
        b2[0] = half ? 1.0f : s0h;    // bias row
        b2[1] = half ? 0.0f : s1h;

        v8f o = {};
        o = __builtin_amdgcn_wmma_f32_16x16x4_f32(false, aout[0], false, b0,
                                                  (short)0, o, false, false);
        o = __builtin_amdgcn_wmma_f32_16x16x4_f32(false, aout[1], false, b1,
                                                  (short)0, o, false, false);
        o = __builtin_amdgcn_wmma_f32_16x16x4_f32(false, aout[2], false, b2,
                                                  (short)0, o, false, false);

#pragma unroll
        for (int r = 0; r < 8; ++r) {
            const int oo = r + 8 * half;
            if (oo < 10)
                out[(size_t)(m0 + lq) * 10 + oo] = o[r];
        }
    }
}

extern "C" void kernel_launch(void* const* d_in, const int* in_sizes, int n_in,
                              void* d_out, int out_size, void* d_ws, size_t ws_size,
                              hipStream_t stream) {
    const float* inputs = (const float*)d_in[0];   // [8192, 2048, 2]
    const float* A      = (const float*)d_in[1];   // [10, 10]
    const float* W_in   = (const float*)d_in[2];   // [10, 2]
    const float* b_mod  = (const float*)d_in[3];   // [10]
    const float* lin_W  = (const float*)d_in[4];   // [10, 10]
    const float* lin_b  = (const float*)d_in[5];   // [10]
    float* out  = (float*)d_out;                   // [8192, 10]
    float* Bmat = (float*)d_ws;                    // 100 floats scratch

    expm_skew10_kernel<<<1, 128, 0, stream>>>(A, Bmat);

    const int waves  = RNN_B / 16;        // 512 waves, 16 batch rows each
    const int blocks = waves / 4;         // 4 waves (128 threads) per block
    exprnn_scan_kernel<<<blocks, 128, 0, stream>>>(inputs, Bmat, W_in, b_mod,
                                                   lin_W, lin_b, out);
}